// GCN_33741263078295
// MI455X (gfx1250) — compile-verified
//
#include <hip/hip_runtime.h>
#include <hip/hip_bf16.h>

typedef __attribute__((ext_vector_type(2))) float v2f;
typedef __attribute__((ext_vector_type(8))) float v8f;

#define DH 128  // hidden dim (both layers' output width)

// ---------------- degree / norm kernels ----------------

__global__ void init_deg_kernel(float* __restrict__ deg, int n) {
    int i = blockIdx.x * blockDim.x + threadIdx.x;
    if (i < n) deg[i] = 1.0f;  // self-loop
}

__global__ void deg_kernel(const int* __restrict__ dst, float* __restrict__ deg, int ne) {
    int i = blockIdx.x * blockDim.x + threadIdx.x;
    if (i < ne) atomicAdd(&deg[dst[i]], 1.0f);
}

__global__ void rsqrt_kernel(float* __restrict__ deg, int n) {
    int i = blockIdx.x * blockDim.x + threadIdx.x;
    if (i < n) deg[i] = rsqrtf(deg[i]);
}

__global__ void zero_kernel(float* __restrict__ p, int n) {
    int i = blockIdx.x * blockDim.x + threadIdx.x;
    if (i < n) p[i] = 0.0f;
}

// ---------------- dense transform: H = X @ W via fp32 WMMA ----------------
// Block = 256 threads = 8 waves. Each block computes a 16 x 128 tile of H.
// X tile (16 x DIN) staged in LDS with padded stride (bank-conflict free),
// each wave owns one 16x16 N-tile, K swept in steps of 4 with
// V_WMMA_F32_16X16X4_F32 (fp32 in / fp32 accumulate -> matches reference).
template <int DIN>
__global__ __launch_bounds__(256) void gemm_wmma_f32(const float* __restrict__ X,
                                                     const float* __restrict__ W,
                                                     float* __restrict__ H, int nrows) {
    constexpr int LSTR = DIN + 4;  // padded LDS row stride (floats): kills bank conflicts
    __shared__ float lx[16 * LSTR];

    const int m0 = blockIdx.x * 16;

    // stage 16 rows of X (row-major contiguous -> fully coalesced)
    for (int i = threadIdx.x; i < 16 * DIN; i += 256) {
        int r = i / DIN;
        int c = i - r * DIN;
        lx[r * LSTR + c] = X[(size_t)m0 * DIN + i];
    }
    __syncthreads();

    const int wave = threadIdx.x >> 5;   // 0..7 -> N tile
    const int lane = threadIdx.x & 31;
    const int n0   = wave * 16;
    const int r    = lane & 15;          // A: row M / B,D: column N
    const int kk   = (lane >> 4) << 1;   // 0 for lanes 0-15, 2 for lanes 16-31

    v8f acc = {};
    #pragma unroll 4
    for (int k0 = 0; k0 < DIN; k0 += 4) {
        v2f a, b;
        // A 16x4 f32: lanes 0-15 K=k0+0/k0+1 ; lanes 16-31 K=k0+2/k0+3
        a.x = lx[r * LSTR + k0 + kk + 0];
        a.y = lx[r * LSTR + k0 + kk + 1];
        // B 4x16 f32: mirrored layout, N striped across lanes
        b.x = W[(size_t)(k0 + kk + 0) * DH + n0 + r];
        b.y = W[(size_t)(k0 + kk + 1) * DH + n0 + r];
        acc = __builtin_amdgcn_wmma_f32_16x16x4_f32(
            /*neg_a=*/false, a, /*neg_b=*/false, b,
            /*c_mod=*/(short)0, acc, /*reuse_a=*/false, /*reuse_b=*/false);
    }

    // D layout: VGPR t -> M = t (lanes 0-15) / M = 8 + t (lanes 16-31)
    const int mbase = m0 + ((lane >> 4) << 3);
    const int col   = n0 + r;
    #pragma unroll
    for (int t = 0; t < 8; ++t) {
        H[(size_t)(mbase + t) * DH + col] = acc[t];
    }
}

// ---------------- edge scatter: AGG[dst] += H[src] * dinv[src]*dinv[dst] ----------------
// One wave32 per edge; 128 floats/row = exactly one float4 per lane.
__global__ __launch_bounds__(256) void scatter_kernel(const float* __restrict__ H,
                                                      const int* __restrict__ src,
                                                      const int* __restrict__ dst,
                                                      const float* __restrict__ dinv,
                                                      float* __restrict__ agg, int ne) {
    const int e = blockIdx.x * 8 + (threadIdx.x >> 5);
    if (e >= ne) return;
    const int lane = threadIdx.x & 31;
    const int s = src[e];
    const int d = dst[e];
    const float w = dinv[s] * dinv[d];

    float4 v = ((const float4*)(H + (size_t)s * DH))[lane];  // 512B coalesced per wave
    float* o = agg + (size_t)d * DH + lane * 4;
    atomicAdd(o + 0, v.x * w);
    atomicAdd(o + 1, v.y * w);
    atomicAdd(o + 2, v.z * w);
    atomicAdd(o + 3, v.w * w);
}

// ---------------- epilogue: OUT = relu(AGG + H * dinv^2 (self-loop) + b) ----------------
__global__ void epilogue_kernel(const float* __restrict__ agg, const float* __restrict__ H,
                                const float* __restrict__ dinv, const float* __restrict__ bias,
                                float* __restrict__ out, int total) {
    int i = blockIdx.x * blockDim.x + threadIdx.x;
    if (i >= total) return;
    int node = i >> 7;   // /128
    int j    = i & 127;
    float di = dinv[node];
    float v  = agg[i] + H[i] * di * di + bias[j];
    out[i] = fmaxf(v, 0.0f);
}

// ---------------- launcher ----------------

extern "C" void kernel_launch(void* const* d_in, const int* in_sizes, int n_in,
                              void* d_out, int out_size, void* d_ws, size_t ws_size,
                              hipStream_t stream) {
    const float* x  = (const float*)d_in[0];
    const int*   ei = (const int*)d_in[1];   // [2, E] int32 (JAX x64 off)
    const float* W1 = (const float*)d_in[2];
    const float* b1 = (const float*)d_in[3];
    const float* W2 = (const float*)d_in[4];
    const float* b2 = (const float*)d_in[5];
    float* out = (float*)d_out;

    const int n  = in_sizes[0] / 256;  // 100000 nodes (Din = 256)
    const int ne = in_sizes[1] / 2;    // 1.6M edges
    const int* srcp = ei;
    const int* dstp = ei + ne;

    const size_t nd = (size_t)n * DH;  // 12.8M floats

    // workspace layout (f32): dinv | H | AGG | L1
    float* dinv = (float*)d_ws;
    float* H    = dinv + (((size_t)n + 255) / 256) * 256;
    float* AGG  = H + nd;
    float* L1   = AGG + nd;

    const int T = 256;
    const int nb_nodes = (n + T - 1) / T;
    const int nb_edges = (ne + T - 1) / T;
    const int nb_feat  = (int)((nd + T - 1) / T);
    const int nb_scat  = (ne + 7) / 8;      // one wave per edge, 8 waves/block
    const int nb_gemm  = n / 16;            // 100000 % 16 == 0

    // degrees -> dinv
    init_deg_kernel<<<nb_nodes, T, 0, stream>>>(dinv, n);
    deg_kernel<<<nb_edges, T, 0, stream>>>(dstp, dinv, ne);
    rsqrt_kernel<<<nb_nodes, T, 0, stream>>>(dinv, n);

    // ---- layer 1 ----
    gemm_wmma_f32<256><<<nb_gemm, T, 0, stream>>>(x, W1, H, n);
    zero_kernel<<<nb_feat, T, 0, stream>>>(AGG, (int)nd);
    scatter_kernel<<<nb_scat, T, 0, stream>>>(H, srcp, dstp, dinv, AGG, ne);
    epilogue_kernel<<<nb_feat, T, 0, stream>>>(AGG, H, dinv, b1, L1, (int)nd);

    // ---- layer 2 (reuse H and AGG) ----
    gemm_wmma_f32<128><<<nb_gemm, T, 0, stream>>>(L1, W2, H, n);
    zero_kernel<<<nb_feat, T, 0, stream>>>(AGG, (int)nd);
    scatter_kernel<<<nb_scat, T, 0, stream>>>(H, srcp, dstp, dinv, AGG, ne);
    epilogue_kernel<<<nb_feat, T, 0, stream>>>(AGG, H, dinv, b2, out, (int)nd);
}